// PiNet2_32074815767161
// MI455X (gfx1250) — compile-verified
//
#include <hip/hip_runtime.h>
#include <hip/hip_bf16.h>

#if defined(__HIP_DEVICE_COMPILE__)
#if !__has_builtin(__builtin_amdgcn_wmma_f32_16x16x4_f32)
#error "missing __builtin_amdgcn_wmma_f32_16x16x4_f32 on device toolchain"
#endif
#endif

typedef float v2f __attribute__((ext_vector_type(2)));
typedef float v8f __attribute__((ext_vector_type(8)));

#define N_ATOMS 20000
#define N_PAIRS 640000
#define ST68    68      // padded LDS row stride (floats) -> conflict-free frag reads
#define PAIR_WAVES 8
#define PW_FLOATS 3456  // per-wave LDS scratch (floats)

// D = A(16x4,f32) * B(4x16,f32) + C ; exact fp32 on the matrix pipe
static __device__ __forceinline__ v8f wmma4(v2f a, v2f b, v8f c) {
#if defined(__HIP_DEVICE_COMPILE__)
  return __builtin_amdgcn_wmma_f32_16x16x4_f32(false, a, false, b, (short)0, c,
                                               false, false);
#else
  (void)a; (void)b;
  return c;
#endif
}
static __device__ __forceinline__ v8f vzero8() {
  v8f z = {0.f, 0.f, 0.f, 0.f, 0.f, 0.f, 0.f, 0.f};
  return z;
}

// Branchless tanh: 1 - 2/(exp(2x)+1).  v_exp_f32 + v_rcp_f32, no EXEC
// divergence, correct saturation at +/-inf, NaN propagates.
static __device__ __forceinline__ float fast_tanh(float x) {
#if defined(__HIP_DEVICE_COMPILE__)
  const float e = __expf(2.0f * x);
  return 1.0f - 2.0f * __builtin_amdgcn_rcpf(e + 1.0f);
#else
  return x;
#endif
}

// ---------------------------------------------------------------------------
__global__ void __launch_bounds__(256) zero_f32(float* __restrict__ p, int n) {
  int i = blockIdx.x * blockDim.x + threadIdx.x;
  const int stride = gridDim.x * blockDim.x;
  for (; i < n; i += stride) p[i] = 0.0f;
}

// ---------------------------------------------------------------------------
// out = act( act(in @ W1 + b1) @ W2 + b2 ), 16-row tiles per wave.
// BIAS/ACT are compile-time -> straight-line epilogues.
template <bool BIAS, bool ACT>
__global__ void __launch_bounds__(256) atom_mlp(
    const float* __restrict__ in, const float* __restrict__ W1,
    const float* __restrict__ b1, const float* __restrict__ W2,
    const float* __restrict__ b2, float* __restrict__ out, int rows) {
  extern __shared__ float sm[];
  const int lane = threadIdx.x & 31;
  const int wv = threadIdx.x >> 5;
  const int m = lane & 15;   // column within 16x16 tile / A-row
  const int g = lane >> 4;   // lane group
  const int kb = 2 * g;      // K sub-offset for A/B fragments
  float* st = sm + wv * (16 * ST68);
  const int NT = rows >> 4;

  for (int tile = blockIdx.x * 8 + wv; tile < NT; tile += gridDim.x * 8) {
    const int r0 = tile << 4;
    // ---- GEMM1: (16x64) @ W1(64x64)
    v8f acc[4];
#pragma unroll
    for (int n = 0; n < 4; ++n) acc[n] = vzero8();
    for (int k = 0; k < 16; ++k) {
      const int kk = 4 * k + kb;
      v2f a = *(const v2f*)(in + (r0 + m) * 64 + kk);
#pragma unroll
      for (int n = 0; n < 4; ++n) {
        v2f b;
        b.x = W1[kk * 64 + 16 * n + m];
        b.y = W1[(kk + 1) * 64 + 16 * n + m];
        acc[n] = wmma4(a, b, acc[n]);
      }
    }
#pragma unroll
    for (int n = 0; n < 4; ++n)
#pragma unroll
      for (int vv = 0; vv < 8; ++vv) {
        const int col = 16 * n + m, row = vv + 8 * g;
        float val = acc[n][vv];
        if (BIAS) val += b1[col];
        if (ACT) val = fast_tanh(val);
        st[row * ST68 + col] = val;   // stage for transposed A reads
      }
    // ---- GEMM2: (16x64) @ W2(64x64)
    v8f ac2[4];
#pragma unroll
    for (int n = 0; n < 4; ++n) ac2[n] = vzero8();
    for (int k = 0; k < 16; ++k) {
      const int kk = 4 * k + kb;
      v2f a = *(const v2f*)(st + m * ST68 + kk);
#pragma unroll
      for (int n = 0; n < 4; ++n) {
        v2f b;
        b.x = W2[kk * 64 + 16 * n + m];
        b.y = W2[(kk + 1) * 64 + 16 * n + m];
        ac2[n] = wmma4(a, b, ac2[n]);
      }
    }
#pragma unroll
    for (int n = 0; n < 4; ++n)
#pragma unroll
      for (int vv = 0; vv < 8; ++vv) {
        const int col = 16 * n + m, row = vv + 8 * g;
        float val = ac2[n][vv];
        if (BIAS) val += b2[col];
        if (ACT) val = fast_tanh(val);
        out[(r0 + row) * 64 + col] = val;
      }
  }
}

// ---------------------------------------------------------------------------
// Fused per-pair pipeline. Weights live in WGP LDS (~193 KB of 320 KB);
// each wave owns one 16-pair tile at a time, everything else in registers.
__global__ void __launch_bounds__(256) pair_kernel(
    const int* __restrict__ ind2, const float* __restrict__ diff,
    const float* __restrict__ basis, const float* __restrict__ Wpi1,
    const float* __restrict__ bpi1, const float* __restrict__ Wpi2,
    const float* __restrict__ bpi2, const float* __restrict__ Wii1,
    const float* __restrict__ Wii2, const float* __restrict__ Wpxi,
    const float* __restrict__ Wpxj, const float* __restrict__ p1h,
    const float* __restrict__ p3h, float* __restrict__ p1n,
    float* __restrict__ p3n) {
  extern __shared__ float sm[];
  float* sWpi1 = sm;               // 128x64
  float* sWpi2 = sWpi1 + 8192;     // 64x256
  float* sWii1 = sWpi2 + 16384;    // 64x64
  float* sWii2 = sWii1 + 4096;     // 64x192
  float* sWpxi = sWii2 + 12288;    // 64x64
  float* sWpxj = sWpxi + 4096;     // 64x64
  float* sbpi1 = sWpxj + 4096;     // 64
  float* sbpi2 = sbpi1 + 64;       // 256
  float* wavebase = sbpi2 + 256;

  const int tid = threadIdx.x;
  for (int i = tid; i < 8192; i += 256) sWpi1[i] = Wpi1[i];
  for (int i = tid; i < 16384; i += 256) sWpi2[i] = Wpi2[i];
  for (int i = tid; i < 4096; i += 256) sWii1[i] = Wii1[i];
  for (int i = tid; i < 12288; i += 256) sWii2[i] = Wii2[i];
  for (int i = tid; i < 4096; i += 256) {
    sWpxi[i] = Wpxi[i];
    sWpxj[i] = Wpxj[i];
  }
  if (tid < 64) sbpi1[tid] = bpi1[tid];
  if (tid < 256) sbpi2[tid] = bpi2[tid];
  __syncthreads();

  const int lane = tid & 31;
  const int wv = tid >> 5;
  const int m = lane & 15;
  const int g = lane >> 4;
  const int kb = 2 * g;

  float* stA = wavebase + wv * PW_FLOATS;  // 16 x ST68 activation tile
  float* st11 = stA + 16 * ST68;           // i1_1 tile
  float* st13 = st11 + 16 * ST68;          // i1 / i1_3 tile
  int* sIdx = (int*)(st13 + 16 * ST68);    // 16 i-idx + 16 j-idx
  float* sBas = (float*)(sIdx + 32);       // 16x4
  float* sDif = sBas + 64;                 // 16x3

  const int NT = N_PAIRS / 16;
  for (int tile = blockIdx.x * PAIR_WAVES + wv; tile < NT;
       tile += gridDim.x * PAIR_WAVES) {
    const int p0 = tile * 16;
    if (lane < 16) {
      const int p = p0 + lane;
      sIdx[lane] = ind2[2 * p];
      sIdx[16 + lane] = ind2[2 * p + 1];
      sBas[lane * 4 + 0] = basis[p * 4 + 0];
      sBas[lane * 4 + 1] = basis[p * 4 + 1];
      sBas[lane * 4 + 2] = basis[p * 4 + 2];
      sBas[lane * 4 + 3] = basis[p * 4 + 3];
      sDif[lane * 3 + 0] = diff[p * 3 + 0];
      sDif[lane * 3 + 1] = diff[p * 3 + 1];
      sDif[lane * 3 + 2] = diff[p * 3 + 2];
    }
    const int myI = sIdx[m], myJ = sIdx[16 + m];
    int rowI[8];
    float basR[8];
#pragma unroll
    for (int vv = 0; vv < 8; ++vv) {
      const int r = vv + 8 * g;
      rowI[vv] = sIdx[r];
      basR[vv] = sBas[r * 4 + (lane & 3)];
    }

    // ---- inter1 = tanh([p1h[i]|p1h[j]] @ Wpi1 + b)  (K=128, N=64)
    v8f acc[4];
#pragma unroll
    for (int n = 0; n < 4; ++n) acc[n] = vzero8();
    for (int k = 0; k < 32; ++k) {
      const float* ab =
          (k < 16) ? (p1h + myI * 64 + 4 * k) : (p1h + myJ * 64 + 4 * k - 64);
      v2f a = *(const v2f*)(ab + kb);
      const int kk = 4 * k + kb;
#pragma unroll
      for (int n = 0; n < 4; ++n) {
        v2f b;
        b.x = sWpi1[kk * 64 + 16 * n + m];
        b.y = sWpi1[(kk + 1) * 64 + 16 * n + m];
        acc[n] = wmma4(a, b, acc[n]);
      }
    }
#pragma unroll
    for (int n = 0; n < 4; ++n)
#pragma unroll
      for (int vv = 0; vv < 8; ++vv) {
        const int col = 16 * n + m;
        stA[(vv + 8 * g) * ST68 + col] = fast_tanh(acc[n][vv] + sbpi1[col]);
      }

    // ---- inter2 = tanh(inter1 @ Wpi2 + b) (N=256), fused basis contraction
    for (int nc = 0; nc < 4; ++nc) {
      v8f a2[4];
#pragma unroll
      for (int n = 0; n < 4; ++n) a2[n] = vzero8();
      for (int k = 0; k < 16; ++k) {
        const int kk = 4 * k + kb;
        v2f a = *(const v2f*)(stA + m * ST68 + kk);
#pragma unroll
        for (int n2 = 0; n2 < 4; ++n2) {
          const int n = nc * 4 + n2;
          v2f b;
          b.x = sWpi2[kk * 256 + 16 * n + m];
          b.y = sWpi2[(kk + 1) * 256 + 16 * n + m];
          a2[n2] = wmma4(a, b, a2[n2]);
        }
      }
#pragma unroll
      for (int n2 = 0; n2 < 4; ++n2) {
        const int n = nc * 4 + n2;
#pragma unroll
        for (int vv = 0; vv < 8; ++vv) {
          float val = fast_tanh(a2[n2][vv] + sbpi2[16 * n + m]);
          val *= basR[vv];                 // * basis[p, col%4]
          val += __shfl_xor(val, 1);       // sum over 4 adjacent columns
          val += __shfl_xor(val, 2);
          if ((m & 3) == 0)
            st13[(vv + 8 * g) * ST68 + 4 * n + (m >> 2)] = val;  // i1
        }
      }
    }

    // ---- i1a = tanh(i1 @ Wii1)
    v8f a3[4];
#pragma unroll
    for (int n = 0; n < 4; ++n) a3[n] = vzero8();
    for (int k = 0; k < 16; ++k) {
      const int kk = 4 * k + kb;
      v2f a = *(const v2f*)(st13 + m * ST68 + kk);
#pragma unroll
      for (int n = 0; n < 4; ++n) {
        v2f b;
        b.x = sWii1[kk * 64 + 16 * n + m];
        b.y = sWii1[(kk + 1) * 64 + 16 * n + m];
        a3[n] = wmma4(a, b, a3[n]);
      }
    }
#pragma unroll
    for (int n = 0; n < 4; ++n)
#pragma unroll
      for (int vv = 0; vv < 8; ++vv)
        stA[(vv + 8 * g) * ST68 + 16 * n + m] = fast_tanh(a3[n][vv]);

    // ---- i1b = tanh(i1a @ Wii2) (N=192): i1_1 -> LDS, i1_2 -> atomic, i1_3 -> LDS
    for (int nc = 0; nc < 3; ++nc) {
      v8f a4[4];
#pragma unroll
      for (int n = 0; n < 4; ++n) a4[n] = vzero8();
      for (int k = 0; k < 16; ++k) {
        const int kk = 4 * k + kb;
        v2f a = *(const v2f*)(stA + m * ST68 + kk);
#pragma unroll
        for (int n2 = 0; n2 < 4; ++n2) {
          const int n = nc * 4 + n2;
          v2f b;
          b.x = sWii2[kk * 192 + 16 * n + m];
          b.y = sWii2[(kk + 1) * 192 + 16 * n + m];
          a4[n2] = wmma4(a, b, a4[n2]);
        }
      }
#pragma unroll
      for (int n2 = 0; n2 < 4; ++n2)
#pragma unroll
        for (int vv = 0; vv < 8; ++vv) {
          const int row = vv + 8 * g, col = 16 * n2 + m;
          float val = fast_tanh(a4[n2][vv]);
          if (nc == 0)
            st11[row * ST68 + col] = val;
          else if (nc == 1)
            atomicAdd(p1n + rowI[vv] * 64 + col, val);  // segment_sum i1_2
          else
            st13[row * ST68 + col] = val;
        }
    }

    // ---- i3 = p3h[i]@Wpxi + p3h[j]@Wpxj; *i1_3 + diff*i1_1; scatter-add
    for (int x = 0; x < 3; ++x) {
      v8f a5[4];
#pragma unroll
      for (int n = 0; n < 4; ++n) a5[n] = vzero8();
      for (int k = 0; k < 16; ++k) {
        const int kk = 4 * k + kb;
        v2f ai = *(const v2f*)(p3h + (myI * 3 + x) * 64 + kk);
        v2f aj = *(const v2f*)(p3h + (myJ * 3 + x) * 64 + kk);
#pragma unroll
        for (int n = 0; n < 4; ++n) {
          v2f bi, bj;
          bi.x = sWpxi[kk * 64 + 16 * n + m];
          bi.y = sWpxi[(kk + 1) * 64 + 16 * n + m];
          a5[n] = wmma4(ai, bi, a5[n]);
          bj.x = sWpxj[kk * 64 + 16 * n + m];
          bj.y = sWpxj[(kk + 1) * 64 + 16 * n + m];
          a5[n] = wmma4(aj, bj, a5[n]);
        }
      }
#pragma unroll
      for (int n = 0; n < 4; ++n)
#pragma unroll
        for (int vv = 0; vv < 8; ++vv) {
          const int row = vv + 8 * g, col = 16 * n + m;
          float val = a5[n][vv] * st13[row * ST68 + col] +
                      sDif[row * 3 + x] * st11[row * ST68 + col];
          atomicAdd(p3n + (rowI[vv] * 3 + x) * 64 + col, val);
        }
    }
  }
}

// ---------------------------------------------------------------------------
// dot = sum_x (p3n@Wdi)*(p3n@Wdj); p1t1 = dot + p1n; p3t1 = p3n * p1t1
__global__ void __launch_bounds__(256) finalize_k(
    const float* __restrict__ p1n, const float* __restrict__ p3n,
    const float* __restrict__ Wdi, const float* __restrict__ Wdj,
    float* __restrict__ out) {
  const int lane = threadIdx.x & 31;
  const int wv = threadIdx.x >> 5;
  const int m = lane & 15;
  const int g = lane >> 4;
  const int kb = 2 * g;
  float* outP1 = out;
  float* outP3 = out + (size_t)N_ATOMS * 64;
  const int NT = N_ATOMS / 16;

  for (int tile = blockIdx.x * 8 + wv; tile < NT; tile += gridDim.x * 8) {
    const int a0 = tile * 16;
    v8f dacc[4];
#pragma unroll
    for (int n = 0; n < 4; ++n) dacc[n] = vzero8();
    for (int x = 0; x < 3; ++x) {
      v8f u[4], v[4];
#pragma unroll
      for (int n = 0; n < 4; ++n) {
        u[n] = vzero8();
        v[n] = vzero8();
      }
      for (int k = 0; k < 16; ++k) {
        const int kk = 4 * k + kb;
        v2f a = *(const v2f*)(p3n + ((a0 + m) * 3 + x) * 64 + kk);
#pragma unroll
        for (int n = 0; n < 4; ++n) {
          v2f bu, bv;
          bu.x = Wdi[kk * 64 + 16 * n + m];
          bu.y = Wdi[(kk + 1) * 64 + 16 * n + m];
          u[n] = wmma4(a, bu, u[n]);
          bv.x = Wdj[kk * 64 + 16 * n + m];
          bv.y = Wdj[(kk + 1) * 64 + 16 * n + m];
          v[n] = wmma4(a, bv, v[n]);
        }
      }
#pragma unroll
      for (int n = 0; n < 4; ++n) dacc[n] += u[n] * v[n];
    }
#pragma unroll
    for (int n = 0; n < 4; ++n)
#pragma unroll
      for (int vv = 0; vv < 8; ++vv) {
        const int row = vv + 8 * g, atom = a0 + row, col = 16 * n + m;
        const float p1t = dacc[n][vv] + p1n[atom * 64 + col];
        outP1[atom * 64 + col] = p1t;
#pragma unroll
        for (int x = 0; x < 3; ++x)
          outP3[(atom * 3 + x) * 64 + col] =
              p3n[(atom * 3 + x) * 64 + col] * p1t;
      }
  }
}

// ---------------------------------------------------------------------------
extern "C" void kernel_launch(void* const* d_in, const int* in_sizes, int n_in,
                              void* d_out, int out_size, void* d_ws,
                              size_t ws_size, hipStream_t stream) {
  (void)in_sizes; (void)n_in; (void)out_size; (void)ws_size;
  const int* ind2 = (const int*)d_in[0];
  const float* p1 = (const float*)d_in[1];
  const float* p3 = (const float*)d_in[2];
  const float* diff = (const float*)d_in[3];
  const float* basis = (const float*)d_in[4];
  const float* Wpp1a = (const float*)d_in[5];
  const float* bpp1a = (const float*)d_in[6];
  const float* Wpp1b = (const float*)d_in[7];
  const float* bpp1b = (const float*)d_in[8];
  const float* Wpi1 = (const float*)d_in[9];
  const float* bpi1 = (const float*)d_in[10];
  const float* Wpi2 = (const float*)d_in[11];
  const float* bpi2 = (const float*)d_in[12];
  const float* Wii1 = (const float*)d_in[13];
  const float* Wii2 = (const float*)d_in[14];
  const float* Wpp3a = (const float*)d_in[15];
  const float* Wpp3b = (const float*)d_in[16];
  const float* Wpxi = (const float*)d_in[17];
  const float* Wpxj = (const float*)d_in[18];
  const float* Wdi = (const float*)d_in[19];
  const float* Wdj = (const float*)d_in[20];

  float* ws = (float*)d_ws;
  float* p1h = ws;
  float* p3h = p1h + (size_t)N_ATOMS * 64;
  float* p1nw = p3h + (size_t)N_ATOMS * 192;
  float* p3nw = p1nw + (size_t)N_ATOMS * 64;

  zero_f32<<<1024, 256, 0, stream>>>(p1nw, N_ATOMS * (64 + 192));
  atom_mlp<true, true><<<160, 256, 8 * 16 * ST68 * 4, stream>>>(
      p1, Wpp1a, bpp1a, Wpp1b, bpp1b, p1h, N_ATOMS);
  atom_mlp<false, false><<<256, 256, 8 * 16 * ST68 * 4, stream>>>(
      p3, Wpp3a, nullptr, Wpp3b, nullptr, p3h, N_ATOMS * 3);

  const int pairSmem = (49472 + PAIR_WAVES * PW_FLOATS) * 4;  // 308480 B
  (void)hipFuncSetAttribute((const void*)pair_kernel,
                            hipFuncAttributeMaxDynamicSharedMemorySize,
                            pairSmem);
  pair_kernel<<<512, 256, pairSmem, stream>>>(ind2, diff, basis, Wpi1, bpi1,
                                              Wpi2, bpi2, Wii1, Wii2, Wpxi,
                                              Wpxj, p1h, p3h, p1nw, p3nw);
  finalize_k<<<160, 256, 0, stream>>>(p1nw, p3nw, Wdi, Wdj, (float*)d_out);
}